// AutoRegressiveLayer_63376537420380
// MI455X (gfx1250) — compile-verified
//
#include <hip/hip_runtime.h>

typedef __attribute__((ext_vector_type(2))) float v2f;
typedef __attribute__((ext_vector_type(8))) float v8f;

#define B_ROWS 8192
#define DIM    1024
#define TM 64          // batch rows per block
#define TN 128         // output cols per block
#define TK 32          // K slice per LDS stage
#define TKP (TK + 1)   // pad to spread LDS banks

typedef __attribute__((address_space(1))) int g_as1_i32;   // global AS int
typedef __attribute__((address_space(3))) int l_as3_i32;   // LDS AS int

// ---- CDNA5 async global->LDS copy (ASYNCcnt-tracked, no VGPR round trip) ----
__device__ __forceinline__ void async_copy_b32(const float* gsrc, float* ldst) {
#if __has_builtin(__builtin_amdgcn_global_load_async_to_lds_b32)
    // step 1: addrspacecast generic -> AS1 / AS3
    const __attribute__((address_space(1))) float* g1 =
        (const __attribute__((address_space(1))) float*)gsrc;
    __attribute__((address_space(3))) float* l3 =
        (__attribute__((address_space(3))) float*)ldst;
    // step 2: const-strip + pointee reinterpret within same AS
    __builtin_amdgcn_global_load_async_to_lds_b32(
        (g_as1_i32*)g1, (l_as3_i32*)l3, 0, 0);
#else
    unsigned lofs = (unsigned)(size_t)ldst;   // low 32b of generic LDS addr = LDS offset
    asm volatile("global_load_async_to_lds_b32 %0, %1, off"
                 :: "v"(lofs), "v"(gsrc) : "memory");
#endif
}

__device__ __forceinline__ void wait_async0() {
#if __has_builtin(__builtin_amdgcn_s_wait_asynccnt)
    __builtin_amdgcn_s_wait_asynccnt(0);
#else
    asm volatile("s_wait_asynccnt 0x0" ::: "memory");
#endif
}

struct Tiles {
    float A [TM][TKP];   // zp tile (gather of z through perm)
    float Wst[TN][TKP];  // (W_s * tril(-1)) tile
    float Wmt[TN][TKP];  // (W_m * tril(-1)) tile
};

__global__ void zero_logj_kernel(float* __restrict__ p, int n) {
    int i = blockIdx.x * blockDim.x + threadIdx.x;
    if (i < n) p[i] = 0.0f;
}

__global__ __launch_bounds__(256)
void ar_fused_gemm(const float* __restrict__ z,
                   const int*   __restrict__ perm,
                   const float* __restrict__ Ws,
                   const float* __restrict__ bs,
                   const float* __restrict__ Wm,
                   const float* __restrict__ bm,
                   const float* __restrict__ s0,
                   const float* __restrict__ m0,
                   float* __restrict__ x_out,
                   float* __restrict__ logj)
{
    __shared__ Tiles tiles[2];         // double buffer
    __shared__ int   sperm[DIM];       // permutation, staged once

    const int tid    = threadIdx.x;
    const int lane   = tid & 31;
    const int wave   = tid >> 5;
    const int wave_m = wave >> 1;             // 0..3 -> 16-row strip
    const int wave_n = wave &  1;             // 0..1 -> 64-col strip

    const int n0   = blockIdx.x * TN;
    const int row0 = blockIdx.y * TM;
    const int k_end = min(DIM, n0 + TN);      // tril(-1): no K past tile diagonal
    const int nstage = k_end / TK;

    // stage perm into LDS (4KB), reused by every K stage
    #pragma unroll
    for (int i = 0; i < DIM / 256; ++i)
        sperm[i * 256 + tid] = perm[i * 256 + tid];
    __syncthreads();

    const int m_base = wave_m * 16;
    const int n_base = wave_n * 64;
    const int nn15   = lane & 15;
    const int kb     = (lane < 16) ? 0 : 2;   // f32 WMMA fragment K split
    const int mm     = m_base + nn15;

    v8f accS[4], accM[4];
    #pragma unroll
    for (int j = 0; j < 4; ++j) { accS[j] = (v8f){}; accM[j] = (v8f){}; }

    // ---- stage issuer: async raw copies below the diagonal, masked path on it ----
    auto stage_issue = [&](Tiles* t, int k0) {
        // A tile: per-lane gather z[row, perm[k]] straight into LDS
        #pragma unroll
        for (int i = 0; i < (TM * TK) / 256; ++i) {
            int tt = i * 256 + tid;
            int m = tt >> 5, k = tt & 31;
            async_copy_b32(&z[(size_t)(row0 + m) * DIM + sperm[k0 + k]],
                           &t->A[m][k]);
        }
        if (k0 + TK <= n0) {
            // fully below diagonal: mask is all-ones -> raw async copy
            #pragma unroll
            for (int i = 0; i < (TN * TK) / 256; ++i) {
                int tt = i * 256 + tid;
                int n = tt >> 5, k = tt & 31;
                size_t off = (size_t)(n0 + n) * DIM + (k0 + k);
                async_copy_b32(&Ws[off], &t->Wst[n][k]);
                async_copy_b32(&Wm[off], &t->Wmt[n][k]);
            }
        } else {
            // diagonal-intersecting stage: fold tril(-1) mask in at LDS store
            #pragma unroll
            for (int i = 0; i < (TN * TK) / 256; ++i) {
                int tt = i * 256 + tid;
                int n = tt >> 5, k = tt & 31;
                int ng = n0 + n, kg = k0 + k;
                bool keep = kg < ng;
                size_t off = (size_t)ng * DIM + kg;
                t->Wst[n][k] = keep ? Ws[off] : 0.0f;
                t->Wmt[n][k] = keep ? Wm[off] : 0.0f;
            }
        }
    };

    // prologue: fill buffer 0
    stage_issue(&tiles[0], 0);
    wait_async0();
    __syncthreads();

    int cur = 0;
    for (int s = 0; s < nstage; ++s) {
        // prefetch next K slice into the other buffer while we compute
        if (s + 1 < nstage)
            stage_issue(&tiles[cur ^ 1], (s + 1) * TK);

        const Tiles* t = &tiles[cur];
        #pragma unroll
        for (int kk = 0; kk < TK; kk += 4) {
            v2f a;
            a.x = t->A[mm][kk + kb];
            a.y = t->A[mm][kk + kb + 1];
            #pragma unroll
            for (int j = 0; j < 4; ++j) {
                const int nl = n_base + j * 16 + nn15;
                v2f bS, bM;
                bS.x = t->Wst[nl][kk + kb];
                bS.y = t->Wst[nl][kk + kb + 1];
                bM.x = t->Wmt[nl][kk + kb];
                bM.y = t->Wmt[nl][kk + kb + 1];
                accS[j] = __builtin_amdgcn_wmma_f32_16x16x4_f32(
                    false, a, false, bS, (short)0, accS[j], false, false);
                accM[j] = __builtin_amdgcn_wmma_f32_16x16x4_f32(
                    false, a, false, bM, (short)0, accM[j], false, false);
            }
        }

        wait_async0();        // my async writes into tiles[cur^1] have landed
        __syncthreads();      // everyone done reading tiles[cur], writes visible
        cur ^= 1;
    }

    // ---- fused epilogue: bias, tanh, col-0 override, x = zp*exp(s)+m, row sums ----
    const float s0v  = s0[0];
    const float m0v  = m0[0];
    const int   half = lane >> 4;             // C layout: lanes 16..31 hold M = v+8

    float rsum[8];
    #pragma unroll
    for (int v = 0; v < 8; ++v) rsum[v] = 0.0f;

    #pragma unroll
    for (int j = 0; j < 4; ++j) {
        const int n_g   = n0 + n_base + j * 16 + nn15;
        const float bsv = bs[n_g];
        const float bmv = bm[n_g];
        const int   pcol = sperm[n_g];
        #pragma unroll
        for (int v = 0; v < 8; ++v) {
            const int r = row0 + m_base + half * 8 + v;
            float sv = accS[j][v] + bsv;
            float mv = accM[j][v] + bmv;
            sv = (n_g == 0) ? s0v : tanhf(sv);
            mv = (n_g == 0) ? m0v : tanhf(mv);
            const float zpv = z[(size_t)r * DIM + pcol];
            x_out[(size_t)r * DIM + n_g] = zpv * expf(sv) + mv;
            // reduce s over the 16 lanes sharing row r (same half)
            float p = sv;
            p += __shfl_xor(p, 1, 16);
            p += __shfl_xor(p, 2, 16);
            p += __shfl_xor(p, 4, 16);
            p += __shfl_xor(p, 8, 16);
            rsum[v] += p;
        }
    }
    if (nn15 == 0) {
        #pragma unroll
        for (int v = 0; v < 8; ++v) {
            const int r = row0 + m_base + half * 8 + v;
            atomicAdd(&logj[r], rsum[v]);
        }
    }
}

extern "C" void kernel_launch(void* const* d_in, const int* in_sizes, int n_in,
                              void* d_out, int out_size, void* d_ws, size_t ws_size,
                              hipStream_t stream) {
    (void)in_sizes; (void)n_in; (void)out_size; (void)d_ws; (void)ws_size;
    const float* z    = (const float*)d_in[0];
    const int*   perm = (const int*)  d_in[1];
    const float* Ws   = (const float*)d_in[2];
    const float* bs   = (const float*)d_in[3];
    const float* Wm   = (const float*)d_in[4];
    const float* bm   = (const float*)d_in[5];
    const float* s0   = (const float*)d_in[6];
    const float* m0   = (const float*)d_in[7];

    float* x    = (float*)d_out;                    // 8192 x 1024
    float* logj = x + (size_t)B_ROWS * DIM;         // 8192

    zero_logj_kernel<<<(B_ROWS + 255) / 256, 256, 0, stream>>>(logj, B_ROWS);

    dim3 grid(DIM / TN, B_ROWS / TM);               // 8 x 128 blocks
    ar_fused_gemm<<<grid, 256, 0, stream>>>(z, perm, Ws, bs, Wm, bm, s0, m0, x, logj);
}